// GATLinkPred_78134045049232
// MI455X (gfx1250) — compile-verified
//
#include <hip/hip_runtime.h>
#include <hip/hip_bf16.h>

// ---------------------------------------------------------------------------
// GAT (2 layers, heads=2, hid=out=64) + link decode for gfx1250 (CDNA5).
// GEMMs use V_WMMA_F32_16X16X4_F32 (full fp32 precision; workload is
// bandwidth-bound so low-precision WMMA buys nothing on the roofline).
// Edge passes use wave-per-edge coalesced float atomics.
// ---------------------------------------------------------------------------

typedef __attribute__((ext_vector_type(2))) float v2f;
typedef __attribute__((ext_vector_type(8))) float v8f;

#define HEADS 2
#define FDIM 128              // IN_CH == HEADS*HID == HEADS*OUT == 128
#define NEG_SLOPE 0.2f

// -------- order-preserving float<->uint encoding for atomic max ------------
__device__ __forceinline__ unsigned enc_f32(float f) {
    unsigned u = __float_as_uint(f);
    return (u & 0x80000000u) ? ~u : (u | 0x80000000u);
}
__device__ __forceinline__ float dec_f32(unsigned u) {
    return __uint_as_float((u & 0x80000000u) ? (u & 0x7FFFFFFFu) : ~u);
}

// ---------------------------------------------------------------------------
// GEMM: Y[M,128] = X[M,128] @ W[128,128], fp32 WMMA 16x16x4.
// Block = 256 threads = 8 waves; wave w handles rows [blk*128 + w*16, +16).
// W staged in LDS (64 KB). M is a multiple of 16 (50000 = 3125*16).
// ---------------------------------------------------------------------------
__global__ __launch_bounds__(256) void gemm128_wmma(
    const float* __restrict__ X, const float* __restrict__ W,
    float* __restrict__ Y, int M)
{
    __shared__ float lw[FDIM * FDIM];   // 65536 bytes
    {   // cooperative load of W into LDS (float4)
        const float4* w4 = (const float4*)W;
        float4* l4 = (float4*)lw;
        for (int i = threadIdx.x; i < FDIM * FDIM / 4; i += 256) l4[i] = w4[i];
    }
    __syncthreads();

    const int wave = threadIdx.x >> 5;
    const int lane = threadIdx.x & 31;
    const int m0 = (blockIdx.x * 8 + wave) * 16;
    if (m0 >= M) return;

    const int lm = lane & 15;   // row-in-tile (A) / col-in-tile (B,D)
    const int lh = lane >> 4;   // half-wave select
    // A fragment source: lane holds A[m0+lm, k0 + 2*lh .. +1]  (ISA 16x4 f32 layout)
    const float* arow = X + (size_t)(m0 + lm) * FDIM + 2 * lh;

#pragma unroll
    for (int nt = 0; nt < 8; ++nt) {
        v8f acc = {};
        const int nc = nt * 16 + lm;
#pragma unroll 8
        for (int kk = 0; kk < 32; ++kk) {
            const int k0 = kk * 4;
            v2f a = *(const v2f*)(arow + k0);                 // aligned 8B load
            v2f b;
            b.x = lw[(k0 + 2 * lh) * FDIM + nc];              // B[k, n]
            b.y = lw[(k0 + 2 * lh + 1) * FDIM + nc];
            acc = __builtin_amdgcn_wmma_f32_16x16x4_f32(
                false, a, false, b, (short)0, acc, false, false);
        }
        // D layout: VGPR r -> row m0 + r + 8*lh, col nc
        float* yrow = Y + (size_t)(m0 + 8 * lh) * FDIM + nc;
#pragma unroll
        for (int r = 0; r < 8; ++r) yrow[(size_t)r * FDIM] = acc[r];
    }
}

// ---------------------------------------------------------------------------
// Per-(node,head) attention logits: e_src[n,h] = <h[n,h,:], a_src[h,:]>, same
// for e_dst. One thread per (node,head).
// ---------------------------------------------------------------------------
__global__ void attn_logits(const float* __restrict__ h,
                            const float* __restrict__ a_src,
                            const float* __restrict__ a_dst,
                            float* __restrict__ esrc, float* __restrict__ edst,
                            int n)
{
    int idx = blockIdx.x * blockDim.x + threadIdx.x;
    if (idx >= n * HEADS) return;
    int node = idx >> 1, hd = idx & 1;
    const float* hp = h + (size_t)node * FDIM + hd * 64;
    const float* as = a_src + hd * 64;
    const float* ad = a_dst + hd * 64;
    float s = 0.f, t = 0.f;
#pragma unroll 8
    for (int f = 0; f < 64; ++f) { float x = hp[f]; s += x * as[f]; t += x * ad[f]; }
    esrc[idx] = s; edst[idx] = t;
}

// ---------------------------------------------------------------------------
// Zero the per-layer accumulators (agg, denom) and the encoded max.
// ---------------------------------------------------------------------------
__global__ void init_accum(unsigned* __restrict__ m, float* __restrict__ denom,
                           float* __restrict__ agg, int n)
{
    int i = blockIdx.x * blockDim.x + threadIdx.x;
    if (i < n * FDIM) agg[i] = 0.f;
    if (i < n * HEADS) { m[i] = 0u; denom[i] = 0.f; }
}

// ---------------------------------------------------------------------------
// Pass A: segment max of LeakyReLU(e_src[src]+e_dst[dst]) per (dst, head).
// One thread per edge (self-loops appended logically: e >= E -> loop e-E).
// ---------------------------------------------------------------------------
__global__ void edge_max(const int* __restrict__ src, const int* __restrict__ dst,
                         int E, int n,
                         const float* __restrict__ esrc,
                         const float* __restrict__ edst,
                         unsigned* __restrict__ m)
{
    int e = blockIdx.x * blockDim.x + threadIdx.x;
    if (e >= E + n) return;
    int s = (e < E) ? src[e] : (e - E);
    int d = (e < E) ? dst[e] : (e - E);
#pragma unroll
    for (int h = 0; h < HEADS; ++h) {
        float v = esrc[s * HEADS + h] + edst[d * HEADS + h];
        v = (v >= 0.f) ? v : NEG_SLOPE * v;
        atomicMax(&m[d * HEADS + h], enc_f32(v));
    }
}

// ---------------------------------------------------------------------------
// Pass B: alpha = exp(e - max); denom[dst,h] += alpha;
//         agg[dst,h,f] += alpha * h[src,h,f].
// One WAVE (32 lanes) per edge: lane covers features f, f+32, f+64, f+96
// (4 coalesced GLOBAL_ATOMIC_ADD_F32 per lane).
// ---------------------------------------------------------------------------
__global__ void edge_aggregate(const int* __restrict__ src, const int* __restrict__ dst,
                               int E, int n,
                               const float* __restrict__ esrc,
                               const float* __restrict__ edst,
                               const unsigned* __restrict__ m,
                               const float* __restrict__ hfeat,
                               float* __restrict__ denom,
                               float* __restrict__ agg)
{
    int wid = (blockIdx.x * blockDim.x + threadIdx.x) >> 5;   // edge id
    int lane = threadIdx.x & 31;
    if (wid >= E + n) return;
    int s = (wid < E) ? src[wid] : (wid - E);
    int d = (wid < E) ? dst[wid] : (wid - E);

    float v0 = esrc[s * HEADS + 0] + edst[d * HEADS + 0];
    float v1 = esrc[s * HEADS + 1] + edst[d * HEADS + 1];
    v0 = (v0 >= 0.f) ? v0 : NEG_SLOPE * v0;
    v1 = (v1 >= 0.f) ? v1 : NEG_SLOPE * v1;
    float a0 = __expf(v0 - dec_f32(m[d * HEADS + 0]));
    float a1 = __expf(v1 - dec_f32(m[d * HEADS + 1]));

    if (lane == 0) {
        atomicAdd(&denom[d * HEADS + 0], a0);
        atomicAdd(&denom[d * HEADS + 1], a1);
    }
    const float* hs = hfeat + (size_t)s * FDIM;
    float* od = agg + (size_t)d * FDIM;
#pragma unroll
    for (int i = 0; i < 4; ++i) {
        int f = lane + 32 * i;
        float al = (f < 64) ? a0 : a1;
        atomicAdd(&od[f], al * hs[f]);
    }
}

// ---------------------------------------------------------------------------
// Normalize by denom, add bias, optional ReLU.  One thread per (node,feature).
// ---------------------------------------------------------------------------
__global__ void node_finalize(const float* __restrict__ agg,
                              const float* __restrict__ denom,
                              const float* __restrict__ bias,
                              float* __restrict__ z, int n, int do_relu)
{
    int idx = blockIdx.x * blockDim.x + threadIdx.x;
    if (idx >= n * FDIM) return;
    int node = idx >> 7, c = idx & 127, hd = c >> 6;
    float v = agg[idx] / denom[node * HEADS + hd] + bias[c];
    if (do_relu) v = fmaxf(v, 0.f);
    z[idx] = v;
}

// ---------------------------------------------------------------------------
// Link decode: scores[e] = <z[src_e], z[dst_e]>, original edges only.
// One wave per edge + shuffle reduction.
// ---------------------------------------------------------------------------
__global__ void decode_edges(const int* __restrict__ src, const int* __restrict__ dst,
                             const float* __restrict__ z,
                             float* __restrict__ out, int E)
{
    int wid = (blockIdx.x * blockDim.x + threadIdx.x) >> 5;
    int lane = threadIdx.x & 31;
    if (wid >= E) return;
    const float* zs = z + (size_t)src[wid] * FDIM;
    const float* zd = z + (size_t)dst[wid] * FDIM;
    float acc = 0.f;
#pragma unroll
    for (int i = 0; i < 4; ++i) { int f = lane + 32 * i; acc += zs[f] * zd[f]; }
#pragma unroll
    for (int off = 16; off > 0; off >>= 1) acc += __shfl_down(acc, off, 32);
    if (lane == 0) out[wid] = acc;
}

// ---------------------------------------------------------------------------
extern "C" void kernel_launch(void* const* d_in, const int* in_sizes, int n_in,
                              void* d_out, int out_size, void* d_ws, size_t ws_size,
                              hipStream_t stream) {
    const float* x      = (const float*)d_in[0];
    const int*   ei     = (const int*)d_in[1];
    const float* W1     = (const float*)d_in[2];
    const float* a1s    = (const float*)d_in[3];
    const float* a1d    = (const float*)d_in[4];
    const float* b1     = (const float*)d_in[5];
    const float* W2     = (const float*)d_in[6];
    const float* a2s    = (const float*)d_in[7];
    const float* a2d    = (const float*)d_in[8];
    const float* b2     = (const float*)d_in[9];
    float* scores = (float*)d_out;

    const int N = in_sizes[0] / FDIM;     // 50000
    const int E = in_sizes[1] / 2;        // 1600000
    const int* src = ei;
    const int* dst = ei + E;

    // workspace layout (floats)
    float* ws   = (float*)d_ws;
    float*    h     = ws;                              // N*128
    float*    agg   = ws + (size_t)N * FDIM;           // N*128
    float*    z     = ws + (size_t)2 * N * FDIM;       // N*128
    float*    esrc  = ws + (size_t)3 * N * FDIM;       // N*2
    float*    edst  = esrc + (size_t)N * HEADS;        // N*2
    unsigned* mmax  = (unsigned*)(edst + (size_t)N * HEADS); // N*2
    float*    denom = (float*)mmax + (size_t)N * HEADS;      // N*2

    const int ETOT = E + N;
    dim3 blk(256);
    dim3 gGemm((N + 127) / 128);
    dim3 gNH((N * HEADS + 255) / 256);
    dim3 gInit((N * FDIM + 255) / 256);
    dim3 gEdge((ETOT + 255) / 256);
    dim3 gEdgeW((ETOT * 32 + 255) / 256);   // wave per edge
    dim3 gDecW(((size_t)E * 32 + 255) / 256);

    // ---- layer 1 ----
    gemm128_wmma<<<gGemm, blk, 0, stream>>>(x, W1, h, N);
    attn_logits<<<gNH, blk, 0, stream>>>(h, a1s, a1d, esrc, edst, N);
    init_accum<<<gInit, blk, 0, stream>>>(mmax, denom, agg, N);
    edge_max<<<gEdge, blk, 0, stream>>>(src, dst, E, N, esrc, edst, mmax);
    edge_aggregate<<<gEdgeW, blk, 0, stream>>>(src, dst, E, N, esrc, edst, mmax, h, denom, agg);
    node_finalize<<<gInit, blk, 0, stream>>>(agg, denom, b1, z, N, /*relu=*/1);

    // ---- layer 2 ----
    gemm128_wmma<<<gGemm, blk, 0, stream>>>(z, W2, h, N);
    attn_logits<<<gNH, blk, 0, stream>>>(h, a2s, a2d, esrc, edst, N);
    init_accum<<<gInit, blk, 0, stream>>>(mmax, denom, agg, N);
    edge_max<<<gEdge, blk, 0, stream>>>(src, dst, E, N, esrc, edst, mmax);
    edge_aggregate<<<gEdgeW, blk, 0, stream>>>(src, dst, E, N, esrc, edst, mmax, h, denom, agg);
    node_finalize<<<gInit, blk, 0, stream>>>(agg, denom, b2, z, N, /*relu=*/0);

    // ---- decode ----
    decode_edges<<<gDecW, blk, 0, stream>>>(src, dst, z, scores, E);
}